// Attention_29764123361281
// MI455X (gfx1250) — compile-verified
//
#include <hip/hip_runtime.h>
#include <hip/hip_bf16.h>
#include <math.h>

// Problem constants from the reference
#define BB  4
#define TQ  256
#define TK  256
#define DD  1024     // DQ == DK
#define AA  512

typedef float v2f __attribute__((ext_vector_type(2)));
typedef float v8f __attribute__((ext_vector_type(8)));

// ---------------------------------------------------------------------------
// Projection GEMM: P[m,n] = sum_d X[m,d] * W[n,d]
//   X : [M=1024, D=1024] row-major (Q or K flattened)
//   W : [N=512,  D=1024] row-major (torch Linear weight layout)
//   P : [M=1024, N=512]  row-major
// One wave computes a 16(M) x 64(N) tile via V_WMMA_F32_16X16X4_F32,
// looping K = D in steps of 4. A-fragment is reused across the 4 N-subtiles.
// ---------------------------------------------------------------------------
__global__ __launch_bounds__(256)
void proj_wmma_f32_kernel(const float* __restrict__ X,
                          const float* __restrict__ W,
                          float* __restrict__ P)
{
    const int wave = (blockIdx.x * blockDim.x + threadIdx.x) >> 5;  // 0..511
    const int lane = threadIdx.x & 31;

    const int mTile  = wave >> 3;          // 64 M-tiles of 16 rows
    const int nGroup = wave & 7;           // 8 N-groups of 64 cols
    const int m0 = mTile * 16;
    const int n0 = nGroup * 64;

    const int lrow = lane & 15;            // M (for A/C) or N (for B/C) index
    const int khi  = (lane >> 4) << 1;     // upper half-wave holds K+2,K+3

    // Per-lane base pointers (8-byte aligned: k%4==0, khi even)
    const float* aPtr = X + (size_t)(m0 + lrow) * DD + khi;
    const float* bPtr0 = W + (size_t)(n0 +  0 + lrow) * DD + khi;
    const float* bPtr1 = W + (size_t)(n0 + 16 + lrow) * DD + khi;
    const float* bPtr2 = W + (size_t)(n0 + 32 + lrow) * DD + khi;
    const float* bPtr3 = W + (size_t)(n0 + 48 + lrow) * DD + khi;

    v8f c0 = {}, c1 = {}, c2 = {}, c3 = {};

    #pragma unroll 4
    for (int k = 0; k < DD; k += 4) {
        v2f a  = *(const v2f*)(aPtr  + k);
        v2f b0 = *(const v2f*)(bPtr0 + k);
        v2f b1 = *(const v2f*)(bPtr1 + k);
        v2f b2 = *(const v2f*)(bPtr2 + k);
        v2f b3 = *(const v2f*)(bPtr3 + k);
        // 8 args: (neg_a, A, neg_b, B, c_mod, C, reuse_a, reuse_b)
        c0 = __builtin_amdgcn_wmma_f32_16x16x4_f32(false, a, false, b0, (short)0, c0, false, false);
        c1 = __builtin_amdgcn_wmma_f32_16x16x4_f32(false, a, false, b1, (short)0, c1, false, false);
        c2 = __builtin_amdgcn_wmma_f32_16x16x4_f32(false, a, false, b2, (short)0, c2, false, false);
        c3 = __builtin_amdgcn_wmma_f32_16x16x4_f32(false, a, false, b3, (short)0, c3, false, false);
    }

    // C/D layout: VGPR r -> lanes 0-15: M=r, lanes 16-31: M=r+8; N = lane&15
    const int mAdd = (lane >> 4) * 8;
    #pragma unroll
    for (int r = 0; r < 8; ++r) {
        const int row = m0 + r + mAdd;
        float* p = P + (size_t)row * AA + n0 + lrow;
        p[ 0] = c0[r];
        p[16] = c1[r];
        p[32] = c2[r];
        p[48] = c3[r];
    }
}

// ---------------------------------------------------------------------------
// Score + masked softmax. One 256-thread block per (b,q) row.
//   score[t] = sum_a w3[a] * tanh(q_proj[b,q,a] + k_proj[b,t,a])
//   out[t]   = softmax_t( mask ? score : -1e10 )
// q_proj row and w3 staged in LDS; k_proj rows stream from L2.
// Inner reduction: wave32 shuffle; softmax: shared-memory tree.
// ---------------------------------------------------------------------------
__global__ __launch_bounds__(256)
void score_softmax_kernel(const float* __restrict__ qp,   // [B*TQ, A]
                          const float* __restrict__ kp,   // [B*TK, A]
                          const float* __restrict__ w3,   // [A]
                          const int*   __restrict__ mask, // [B*TQ, TK]
                          float* __restrict__ out)        // [B*TQ, TK]
{
    __shared__ float sh_q[AA];
    __shared__ float sh_w[AA];
    __shared__ float sh_score[TK];
    __shared__ float red[256];

    const int row = blockIdx.x;              // 0 .. B*TQ-1
    const int b   = row >> 8;                // TQ = 256
    const int tid = threadIdx.x;

    const float* qrow = qp + (size_t)row * AA;
    for (int i = tid; i < AA; i += 256) {
        sh_q[i] = qrow[i];
        sh_w[i] = w3[i];
    }
    __syncthreads();

    const int wave = tid >> 5;
    const int lane = tid & 31;

    for (int t = wave; t < TK; t += 8) {
        const float* krow = kp + (size_t)(b * TK + t) * AA;
        float acc = 0.0f;
        #pragma unroll 4
        for (int a = lane; a < AA; a += 32) {
            acc = fmaf(sh_w[a], tanhf(sh_q[a] + krow[a]), acc);
        }
        #pragma unroll
        for (int off = 16; off > 0; off >>= 1)
            acc += __shfl_down(acc, off, 32);
        if (lane == 0) sh_score[t] = acc;
    }
    __syncthreads();

    // masked softmax over the 256 scores
    const int* mrow = mask + (size_t)row * TK;
    float s = mrow[tid] ? sh_score[tid] : -1e10f;

    red[tid] = s;
    __syncthreads();
    #pragma unroll
    for (int off = 128; off > 0; off >>= 1) {
        if (tid < off) red[tid] = fmaxf(red[tid], red[tid + off]);
        __syncthreads();
    }
    const float mx = red[0];
    __syncthreads();

    const float e = __expf(s - mx);
    red[tid] = e;
    __syncthreads();
    #pragma unroll
    for (int off = 128; off > 0; off >>= 1) {
        if (tid < off) red[tid] += red[tid + off];
        __syncthreads();
    }
    const float inv = 1.0f / red[0];

    out[(size_t)row * TK + tid] = e * inv;
}

// ---------------------------------------------------------------------------
extern "C" void kernel_launch(void* const* d_in, const int* in_sizes, int n_in,
                              void* d_out, int out_size, void* d_ws, size_t ws_size,
                              hipStream_t stream) {
    const float* Q    = (const float*)d_in[0];  // [B,TQ,1,DQ]  -> [1024,1024]
    const float* K    = (const float*)d_in[1];  // [B,1,TK,DK]  -> [1024,1024]
    const int*   mask = (const int*)  d_in[2];  // [B,TQ,TK]
    const float* W1   = (const float*)d_in[3];  // [A,DQ]
    const float* W2   = (const float*)d_in[4];  // [A,DK]
    const float* w3   = (const float*)d_in[5];  // [A]
    float* out = (float*)d_out;

    float* qp = (float*)d_ws;                   // [B*TQ, A]  2 MB
    float* kp = qp + (size_t)BB * TQ * AA;      // [B*TK, A]  2 MB

    // 512 waves per GEMM -> 64 blocks of 8 waves (256 threads)
    proj_wmma_f32_kernel<<<64, 256, 0, stream>>>(Q, W1, qp);
    proj_wmma_f32_kernel<<<64, 256, 0, stream>>>(K, W2, kp);

    // one block per (b, q) row
    score_softmax_kernel<<<BB * TQ, 256, 0, stream>>>(qp, kp, w3, mask, out);
}